// S4DKernel_84361747628481
// MI455X (gfx1250) — compile-verified
//
#include <hip/hip_runtime.h>
#include <math.h>

// Problem constants (fixed by the reference harness)
#define H_DIM   512
#define N_DIM   64
#define L_DIM   2048
#define CH_DIM  1
#define LTILE   128                 // l-values per block in the Cauchy kernel
#define TWO_PI  6.283185307179586f

typedef __attribute__((ext_vector_type(2))) float v2f;
typedef __attribute__((ext_vector_type(8))) float v8f;

__device__ __forceinline__ float softplusf(float x) {
    return fmaxf(x, 0.0f) + log1pf(expf(-fabsf(x)));
}

// Raw v_rcp_f32 (~1 ulp) instead of the IEEE div expansion.
__device__ __forceinline__ float frcp(float x) {
    return __builtin_amdgcn_rcpf(x);
}

// ---------------------------------------------------------------------------
// Kernel 1: Cauchy resolvent + 4 complex reductions over N via WMMA
//   Per wave: one 16-l tile of one head h. The reduction
//     s_X[l] = sum_n w_X[n] * R[l,n]      (X in {CB, CP, PB, PP})
//   is a real (16 x 128) x (128 x 8) GEMM:
//     A = [Re R | Im R]  (K = 2N = 128)
//     B rows k<64 : col 2X = Re w_X[n],  col 2X+1 = Im w_X[n]
//     B rows k>=64: col 2X = -Im w_X[n], col 2X+1 = Re w_X[n]
//   done as 32 chained V_WMMA_F32_16X16X4_F32.
// ---------------------------------------------------------------------------
__global__ __launch_bounds__(256)
void s4_cauchy_wmma(const float* __restrict__ Lre, const float* __restrict__ Lim,
                    const float* __restrict__ Pre, const float* __restrict__ Pim,
                    const float* __restrict__ Bre, const float* __restrict__ Bim,
                    const float* __restrict__ Cre, const float* __restrict__ Cim,
                    const float* __restrict__ log_dt,
                    float* __restrict__ KhatRe, float* __restrict__ KhatIm)
{
    __shared__ float Wm[128 * 16];          // B-matrix [k][col], cols 8..15 zero
    __shared__ float lamRe[N_DIM], lamIm[N_DIM];
    __shared__ float tile[8 * 16 * 8];      // per-wave 16x8 WMMA result spill

    const int tid    = threadIdx.x;
    const int h      = blockIdx.x / (L_DIM / LTILE);
    const int lchunk = blockIdx.x % (L_DIM / LTILE);

    // Zero the weight matrix (cols 8..15 stay zero)
    for (int i = tid; i < 128 * 16; i += 256) Wm[i] = 0.0f;
    __syncthreads();

    if (tid < N_DIM) {
        const int n   = tid;
        const int idx = h * N_DIM + n;
        const float pr = Pre[idx], pi = Pim[idx];
        const float br = Bre[idx], bi = Bim[idx];
        const float cr = Cre[idx], ci = Cim[idx];      // CH = 1 -> ch index 0
        // conj(P)*B, conj(P)*P, conj(C)*B, conj(C)*P
        const float wPBr = pr * br + pi * bi, wPBi = pr * bi - pi * br;
        const float wPPr = pr * pr + pi * pi, wPPi = 0.0f;
        const float wCBr = cr * br + ci * bi, wCBi = cr * bi - ci * br;
        const float wCPr = cr * pr + ci * pi, wCPi = cr * pi - ci * pr;
        const float wr[4] = {wCBr, wCPr, wPBr, wPPr};
        const float wi[4] = {wCBi, wCPi, wPBi, wPPi};
        #pragma unroll
        for (int X = 0; X < 4; ++X) {
            Wm[n * 16 + 2 * X]              =  wr[X];
            Wm[n * 16 + 2 * X + 1]          =  wi[X];
            Wm[(64 + n) * 16 + 2 * X]       = -wi[X];
            Wm[(64 + n) * 16 + 2 * X + 1]   =  wr[X];
        }
        lamRe[n] = -softplusf(Lre[idx]);
        lamIm[n] = Lim[idx];
    }
    __syncthreads();

    const int wave = tid >> 5;
    const int lane = tid & 31;
    const int half = lane >> 4;             // K-half selector per ISA A layout
    const int c    = lane & 15;             // row m within tile / output column
    const int l0   = lchunk * LTILE + wave * 16;
    const int l    = l0 + c;

    // g[h,l] = (2/dt) * (1-z)/(1+z), z = exp(-2*pi*i*l/L), eps-clamped denom
    const float sc    = 2.0f * expf(-log_dt[h]);
    const float theta = (TWO_PI / (float)L_DIM) * (float)l;
    float zre, zim_p;                        // z = (cos t, -sin t)
    __sincosf(theta, &zim_p, &zre);
    const float zim = -zim_p;
    float dnr = 1.0f + zre, dni = zim;
    float d2  = dnr * dnr + dni * dni;
    const float eps = 1.1920929e-7f;
    if (sqrtf(d2) < eps) { dnr = eps; dni = 0.0f; d2 = eps * eps; }
    const float nur = 1.0f - zre, nui = -zim;
    const float invd2   = frcp(d2);
    const float g_re    = sc * (nur * dnr + nui * dni) * invd2;
    const float g_im    = sc * (nui * dnr - nur * dni) * invd2;
    const float pref_re =  2.0f * dnr * invd2;           // 2 / (1+z)
    const float pref_im = -2.0f * dni * invd2;

    v8f acc = {};
    #pragma unroll
    for (int kk = 0; kk < 16; ++kk) {
        // This lane supplies A[m][k] for k = 4*kk + 2*half + {0,1}
        const int n0  = 4 * kk + 2 * half;               // n index (< 64)
        const float dr0 = g_re - lamRe[n0],     di0 = g_im - lamIm[n0];
        const float dr1 = g_re - lamRe[n0 + 1], di1 = g_im - lamIm[n0 + 1];
        const float iv0 = frcp(dr0 * dr0 + di0 * di0);
        const float iv1 = frcp(dr1 * dr1 + di1 * di1);
        v2f aRe = {  dr0 * iv0,  dr1 * iv1 };            // Re R
        v2f aIm = { -di0 * iv0, -di1 * iv1 };            // Im R
        const int krow = 4 * kk + 2 * half;
        v2f bRe = { Wm[krow * 16 + c],        Wm[(krow + 1) * 16 + c] };
        v2f bIm = { Wm[(64 + krow) * 16 + c], Wm[(64 + krow + 1) * 16 + c] };
        acc = __builtin_amdgcn_wmma_f32_16x16x4_f32(false, aRe, false, bRe,
                                                    (short)0, acc, false, false);
        acc = __builtin_amdgcn_wmma_f32_16x16x4_f32(false, aIm, false, bIm,
                                                    (short)0, acc, false, false);
    }

    // Spill the 8 used output columns: acc[r] holds (M = r + 8*half, N = c)
    if (c < 8) {
        #pragma unroll
        for (int r = 0; r < 8; ++r) {
            const int m = r + 8 * half;
            tile[(wave * 16 + m) * 8 + c] = acc[r];
        }
    }
    __syncthreads();

    // Woodbury epilogue: lanes 0..15 of each wave handle l = l0 + lane
    if (lane < 16) {
        const float* t8 = &tile[(wave * 16 + lane) * 8];
        const float sCBr = t8[0], sCBi = t8[1];
        const float sCPr = t8[2], sCPi = t8[3];
        const float sPBr = t8[4], sPBi = t8[5];
        const float sPPr = t8[6], sPPi = t8[7];
        const float dwr = 1.0f + sPPr, dwi = sPPi;       // 1 + PRP
        const float iw  = frcp(dwr * dwr + dwi * dwi);
        const float tr  = (sPBr * dwr + sPBi * dwi) * iw;   // PRB / (1+PRP)
        const float ti  = (sPBi * dwr - sPBr * dwi) * iw;
        const float cr2 = sCPr * tr - sCPi * ti;            // CRP * (...)
        const float ci2 = sCPr * ti + sCPi * tr;
        const float vr  = sCBr - cr2, vi = sCBi - ci2;      // CRB - correction
        KhatRe[h * L_DIM + l] = pref_re * vr - pref_im * vi;
        KhatIm[h * L_DIM + l] = pref_re * vi + pref_im * vr;
    }
}

// ---------------------------------------------------------------------------
// Kernel 2: length-2048 radix-2 DIT inverse FFT per head in LDS, real part out.
// ---------------------------------------------------------------------------
__global__ __launch_bounds__(256)
void s4_ifft(const float* __restrict__ KhatRe, const float* __restrict__ KhatIm,
             const float* __restrict__ D, float* __restrict__ out)
{
    __shared__ float re[L_DIM];
    __shared__ float im[L_DIM];
    const int tid = threadIdx.x;
    const int h   = blockIdx.x;

    // Bit-reversed load (11-bit reversal)
    for (int i = tid; i < L_DIM; i += 256) {
        const int j = (int)(__brev((unsigned)i) >> 21);
        re[i] = KhatRe[h * L_DIM + j];
        im[i] = KhatIm[h * L_DIM + j];
    }
    __syncthreads();

    for (int s = 1; s <= 11; ++s) {
        const int m  = 1 << s;
        const int hf = m >> 1;
        const float wstep = TWO_PI / (float)m;
        for (int b = tid; b < L_DIM / 2; b += 256) {
            const int j    = b & (hf - 1);
            const int base = ((b >> (s - 1)) << s) + j;
            float wr, wi;
            __sincosf(wstep * (float)j, &wi, &wr);        // +sign: inverse FFT
            const float xr = re[base + hf], xi = im[base + hf];
            const float tr = wr * xr - wi * xi;
            const float ti = wr * xi + wi * xr;
            const float ur = re[base], ui = im[base];
            re[base]      = ur + tr;
            im[base]      = ui + ti;
            re[base + hf] = ur - tr;
            im[base + hf] = ui - ti;
        }
        __syncthreads();
    }

    const float scale = 1.0f / (float)L_DIM;
    for (int i = tid; i < L_DIM; i += 256)
        out[h * L_DIM + i] = re[i] * scale;

    // Append D passthrough (CH*H elements after K); CH==1 so index is h.
    if (tid == 0)
        out[(size_t)CH_DIM * H_DIM * L_DIM + h] = D[h];
}

// ---------------------------------------------------------------------------
extern "C" void kernel_launch(void* const* d_in, const int* in_sizes, int n_in,
                              void* d_out, int out_size, void* d_ws, size_t ws_size,
                              hipStream_t stream)
{
    (void)in_sizes; (void)n_in; (void)out_size; (void)ws_size;
    const float* Lre = (const float*)d_in[0];
    const float* Lim = (const float*)d_in[1];
    const float* Pre = (const float*)d_in[2];
    const float* Pim = (const float*)d_in[3];
    const float* Bre = (const float*)d_in[4];
    const float* Bim = (const float*)d_in[5];
    const float* Cre = (const float*)d_in[6];
    const float* Cim = (const float*)d_in[7];
    const float* D   = (const float*)d_in[8];
    const float* ldt = (const float*)d_in[9];
    // d_in[10] is L (==2048), hardcoded.

    float* out    = (float*)d_out;
    float* KhatRe = (float*)d_ws;                          // H*L floats
    float* KhatIm = KhatRe + (size_t)H_DIM * L_DIM;        // H*L floats (8 MB total)

    s4_cauchy_wmma<<<dim3(H_DIM * (L_DIM / LTILE)), 256, 0, stream>>>(
        Lre, Lim, Pre, Pim, Bre, Bim, Cre, Cim, ldt, KhatRe, KhatIm);
    s4_ifft<<<dim3(H_DIM), 256, 0, stream>>>(KhatRe, KhatIm, D, out);
}